// AdversarialPatch_81673098101467
// MI455X (gfx1250) — compile-verified
//
#include <hip/hip_runtime.h>
#include <stdint.h>
#include <stddef.h>

#define M_BOXES   6144
#define SORT_N    8192
#define MASK_PITCH 192        // 6144/32 words per row
#define RCNN_THRES 0.25f
#define YOLO_THRES 0.45f
#define NMS_THRES  0.4f
#define RATIO      0.8f

#define P_BLOCKS   512
#define R_BLOCKS   24
#define YP_BLOCKS  32

// ---- workspace byte offsets (total ~4.99 MB) ----
#define WS_MASK   ((size_t)0)
#define WS_KEEP   (WS_MASK  + (size_t)M_BOXES*MASK_PITCH*4)   // 4,718,592
#define WS_KEYS   (WS_KEEP  + 768)
#define WS_ORDER  (WS_KEYS  + (size_t)SORT_N*8)
#define WS_X1S    (WS_ORDER + (size_t)M_BOXES*4)
#define WS_Y1S    (WS_X1S   + (size_t)M_BOXES*4)
#define WS_X2S    (WS_Y1S   + (size_t)M_BOXES*4)
#define WS_Y2S    (WS_X2S   + (size_t)M_BOXES*4)
#define WS_AREA   (WS_Y2S   + (size_t)M_BOXES*4)
#define WS_CONFS  (WS_AREA  + (size_t)M_BOXES*4)
#define WS_C5S    (WS_CONFS + (size_t)M_BOXES*4)
#define WS_PPART  (WS_C5S   + (size_t)M_BOXES*4)
#define WS_RPART  (WS_PPART + (size_t)P_BLOCKS*4)
#define WS_BLPART (WS_RPART + (size_t)R_BLOCKS*4)
#define WS_BCPART (WS_BLPART+ (size_t)YP_BLOCKS*4)

typedef float v2f __attribute__((ext_vector_type(2)));
typedef float v8f __attribute__((ext_vector_type(8)));

__device__ __forceinline__ float box_l(float c5, float c4) {
  const float s = 1.0f / (0.5f - YOLO_THRES);   // 20
  float t5 = fminf(fmaxf((c5 - YOLO_THRES) * s, 0.f), 1.f);
  float t4 = fminf(fmaxf((c4 - YOLO_THRES) * s, 0.f), 1.f);
  return -t5 * logf(1.0f - c5 + 0.01f) - t4 * logf(1.0f - c4 + 0.01f);
}

// -------- clear mask + keep words --------
__global__ void clear_k(unsigned* p, int n) {
  int i = blockIdx.x * blockDim.x + threadIdx.x;
  if (i < n) p[i] = 0u;
}

// -------- patch add + out-of-range loss (fixed-order block reduce) --------
__global__ void patch_loss_k(const float* __restrict__ img,
                             const float* __restrict__ p0,
                             const float* __restrict__ p1,
                             const float* __restrict__ p2,
                             float* __restrict__ part) {
  __shared__ float red[256];
  float acc = 0.f;
  int stride = gridDim.x * blockDim.x;
  for (int idx = blockIdx.x * blockDim.x + threadIdx.x; idx < 750000; idx += stride) {
    int c = idx / 250000;
    int rem = idx - c * 250000;
    int y = rem / 500;
    int x = rem - y * 500;
    float v = img[idx];
    if (x >= 50 && x < 450) {
      const float* p = nullptr; int y0 = 0;
      if      (y >=  75 && y < 125) { p = p0; y0 =  75; }
      else if (y >= 225 && y < 275) { p = p1; y0 = 225; }
      else if (y >= 375 && y < 425) { p = p2; y0 = 375; }
      if (p) v += p[c * 20000 + (y - y0) * 400 + (x - 50)];
    }
    acc += fmaxf(-v, 0.f) + fmaxf(v - 1.f, 0.f);
  }
  red[threadIdx.x] = acc; __syncthreads();
  for (int s = 128; s > 0; s >>= 1) {
    if (threadIdx.x < s) red[threadIdx.x] += red[threadIdx.x + s];
    __syncthreads();
  }
  if (threadIdx.x == 0) part[blockIdx.x] = red[0];
}

// -------- RCNN loss: one thread per row of (6144,81) --------
__global__ void rcnn_k(const float* __restrict__ rp, float* __restrict__ part) {
  __shared__ float red[256];
  int i = blockIdx.x * 256 + threadIdx.x;      // always < 6144 (24 blocks)
  const float* row = rp + (size_t)i * 81;
  float prob = row[0];
  #pragma unroll 4
  for (int k = 1; k < 80; ++k) prob = fmaxf(prob, row[k]);
  float acc = 0.f;
  if (prob > RCNN_THRES) {
    float bprob = row[80];
    float cc = fminf(fmaxf((prob - RCNN_THRES) / (0.3f - RCNN_THRES), 0.f), 1.f);
    acc = -logf(bprob + 0.001f) - cc * logf(1.0f - prob + 0.001f);
  }
  red[threadIdx.x] = acc; __syncthreads();
  for (int s = 128; s > 0; s >>= 1) {
    if (threadIdx.x < s) red[threadIdx.x] += red[threadIdx.x + s];
    __syncthreads();
  }
  if (threadIdx.x == 0) part[blockIdx.x] = red[0];
}

// -------- YOLO prep: b_loss/b_cnt partials + 64-bit sort keys --------
__global__ void yolo_prep_k(const float* __restrict__ boxes,
                            unsigned long long* __restrict__ keys,
                            float* __restrict__ blpart, float* __restrict__ bcpart) {
  __shared__ float redl[256], redc[256];
  int i = blockIdx.x * 256 + threadIdx.x;
  float bl = 0.f, bc = 0.f;
  if (i < M_BOXES) {
    float conf = boxes[i * 6 + 4], c5 = boxes[i * 6 + 5];
    unsigned u = __float_as_uint(conf);
    unsigned m = (u & 0x80000000u) ? ~u : (u | 0x80000000u);  // ascending float order
    keys[i] = ((unsigned long long)(~m) << 32) | (unsigned)i; // descending conf, stable tiebreak
    if (conf > YOLO_THRES) { bl = box_l(c5, conf); bc = 1.f; }
  } else if (i < SORT_N) {
    keys[i] = 0xFFFFFFFFFFFFFFFFull;                          // pad sorts to end
  }
  redl[threadIdx.x] = bl; redc[threadIdx.x] = bc; __syncthreads();
  for (int s = 128; s > 0; s >>= 1) {
    if (threadIdx.x < s) { redl[threadIdx.x] += redl[threadIdx.x + s];
                           redc[threadIdx.x] += redc[threadIdx.x + s]; }
    __syncthreads();
  }
  if (threadIdx.x == 0) { blpart[blockIdx.x] = redl[0]; bcpart[blockIdx.x] = redc[0]; }
}

// -------- single-block bitonic sort of 8192 u64 keys in 64KB LDS --------
__global__ void bitonic_k(unsigned long long* __restrict__ keys,
                          unsigned* __restrict__ order) {
  extern __shared__ unsigned long long sk[];
  int tid = threadIdx.x;
  for (int i = tid; i < SORT_N; i += 1024) sk[i] = keys[i];
  __syncthreads();
  for (int k = 2; k <= SORT_N; k <<= 1) {
    for (int j = k >> 1; j > 0; j >>= 1) {
      for (int idx = tid; idx < SORT_N; idx += 1024) {
        int ixj = idx ^ j;
        if (ixj > idx) {
          bool up = ((idx & k) == 0);
          unsigned long long a = sk[idx], b = sk[ixj];
          if ((a > b) == up) { sk[idx] = b; sk[ixj] = a; }
        }
      }
      __syncthreads();
    }
  }
  for (int i = tid; i < M_BOXES; i += 1024)
    order[i] = (unsigned)(sk[i] & 0xFFFFFFFFull);
}

// -------- gather sorted box geometry (SoA) --------
__global__ void gather_k(const float* __restrict__ boxes, const unsigned* __restrict__ order,
                         float* x1s, float* y1s, float* x2s, float* y2s,
                         float* ar, float* confs, float* c5s) {
  int t = blockIdx.x * 256 + threadIdx.x;
  if (t >= M_BOXES) return;
  unsigned o = order[t];
  float bx = boxes[o * 6 + 0], by = boxes[o * 6 + 1];
  float bw = boxes[o * 6 + 2], bh = boxes[o * 6 + 3];
  x1s[t] = bx - bw * 0.5f;  x2s[t] = bx + bw * 0.5f;
  y1s[t] = by - bh * 0.5f;  y2s[t] = by + bh * 0.5f;
  ar[t]  = bw * bh;
  confs[t] = boxes[o * 6 + 4];
  c5s[t]   = boxes[o * 6 + 5];
}

// -------- suppression bitmask: one wave per 16x16 tile; WMMA computes
// the 16x16 outer sum area_i + area_j in one v_wmma_f32_16x16x4_f32 --------
__global__ void __launch_bounds__(32)
nms_mask_k(const float* __restrict__ x1s, const float* __restrict__ y1s,
           const float* __restrict__ x2s, const float* __restrict__ y2s,
           const float* __restrict__ ar, unsigned* __restrict__ mask) {
  int lane = threadIdx.x;
  int rowBase = blockIdx.y * 16;
  int colBase = blockIdx.x * 16;
  if (colBase + 15 <= rowBase) return;      // uniform: no j > i pairs in tile

  __shared__ float rx1[16], ry1[16], rx2[16], ry2[16];
  __shared__ unsigned bsh[8];
  if (lane < 16) {
    rx1[lane] = x1s[rowBase + lane];  ry1[lane] = y1s[rowBase + lane];
    rx2[lane] = x2s[rowBase + lane];  ry2[lane] = y2s[rowBase + lane];
  }
  int c = lane & 15;
  float cx1 = x1s[colBase + c], cy1 = y1s[colBase + c];
  float cx2 = x2s[colBase + c], cy2 = y2s[colBase + c];

  // A(16x4) row m = (area_row[m], 1, 0, 0) ; B(4x16) col n = (1, area_col[n], 0, 0)
  // => D[m][n] = area_row[m] + area_col[n]
  v2f a, b;
  if (lane < 16) { a.x = ar[rowBase + lane]; a.y = 1.0f;
                   b.x = 1.0f;               b.y = ar[colBase + lane]; }
  else           { a.x = 0.f; a.y = 0.f; b.x = 0.f; b.y = 0.f; }   // K=2,3 rows are zero
  v8f cz = {};
  v8f d = __builtin_amdgcn_wmma_f32_16x16x4_f32(false, a, false, b,
                                                (short)0, cz, false, false);
  __syncthreads();

  #pragma unroll
  for (int v = 0; v < 8; ++v) {
    int r = v + ((lane < 16) ? 0 : 8);      // C/D layout: VGPR v -> rows v, v+8
    float iw = fminf(rx2[r], cx2) - fmaxf(rx1[r], cx1);
    float ih = fminf(ry2[r], cy2) - fmaxf(ry1[r], cy1);
    iw = fmaxf(iw, 0.f);  ih = fmaxf(ih, 0.f);
    float inter = iw * ih;
    int gi = rowBase + r, gj = colBase + c;
    // iou > T  <=>  inter*(1+T) > T*(area_i + area_j)
    bool sup = (inter * (1.0f + NMS_THRES) > NMS_THRES * d[v]) && (gj > gi);
    unsigned long long bal = __ballot(sup);
    if (lane == 0) bsh[v] = (unsigned)bal;  // bits[15:0]=row v, bits[31:16]=row v+8
  }
  __syncthreads();
  if (lane < 16) {
    unsigned val = (lane < 8) ? (bsh[lane] & 0xFFFFu) : (bsh[lane - 8] >> 16);
    if (val)
      atomicOr(&mask[(size_t)(rowBase + lane) * MASK_PITCH + (colBase >> 5)],
               val << (colBase & 16));
  }
}

// -------- sequential greedy NMS scan: one wave, keep bits in registers --------
__global__ void __launch_bounds__(32)
nms_scan_k(const unsigned* __restrict__ mask, const float* __restrict__ confs,
           unsigned* __restrict__ keep) {
  int lane = threadIdx.x;
  unsigned K[6];
  #pragma unroll
  for (int reg = 0; reg < 6; ++reg) {       // lane owns word w = reg*32 + lane
    unsigned kw = 0;
    #pragma unroll
    for (int gg = 0; gg < 32; ++gg) {
      int g = reg * 32 + gg;                // word index
      unsigned long long bal = __ballot(confs[g * 32 + lane] > YOLO_THRES);
      if (lane == gg) kw = (unsigned)bal;
    }
    K[reg] = kw;
  }
  #pragma unroll
  for (int seg = 0; seg < 6; ++seg) {       // i in [seg*1024, seg*1024+1024): word>>5 == seg
    #pragma unroll 1
    for (int i = seg * 1024; i < (seg + 1) * 1024; ++i) {
      int w = i >> 5;
      unsigned kw = __shfl(K[seg], w & 31, 32);
      if ((kw >> (i & 31)) & 1u) {
        #pragma unroll
        for (int r = 0; r < 6; ++r)
          K[r] &= ~mask[(size_t)i * MASK_PITCH + (lane + 32 * r)];
      }
    }
  }
  #pragma unroll
  for (int r = 0; r < 6; ++r) keep[lane + 32 * r] = K[r];
}

// -------- final combine --------
__global__ void final_k(const float* __restrict__ ppart, const float* __restrict__ rpart,
                        const float* __restrict__ blpart, const float* __restrict__ bcpart,
                        const unsigned* __restrict__ keep,
                        const float* __restrict__ confs, const float* __restrict__ c5s,
                        float* __restrict__ out) {
  __shared__ float red[256];
  int t = threadIdx.x;
  float nmsl = 0.f, nmsc = 0.f;
  for (int i = t; i < M_BOXES; i += 256) {
    if ((keep[i >> 5] >> (i & 31)) & 1u) { nmsl += box_l(c5s[i], confs[i]); nmsc += 1.f; }
  }
  float pl = ppart[t] + ppart[t + 256];
  float rl = (t < R_BLOCKS)  ? rpart[t]  : 0.f;
  float bl = (t < YP_BLOCKS) ? blpart[t] : 0.f;
  float bc = (t < YP_BLOCKS) ? bcpart[t] : 0.f;

#define BLK_REDUCE(VAR)                                                     \
  red[t] = (VAR); __syncthreads();                                          \
  for (int s = 128; s > 0; s >>= 1) {                                       \
    if (t < s) red[t] += red[t + s];                                        \
    __syncthreads();                                                        \
  }                                                                         \
  (VAR) = red[0]; __syncthreads();

  BLK_REDUCE(nmsl); BLK_REDUCE(nmsc); BLK_REDUCE(pl);
  BLK_REDUCE(rl);   BLK_REDUCE(bl);   BLK_REDUCE(bc);
#undef BLK_REDUCE

  if (t == 0) {
    float yolo = bl + nmsl * (bc / fmaxf(nmsc, 1.0f));
    out[0] = rl * RATIO + yolo + pl;
  }
}

extern "C" void kernel_launch(void* const* d_in, const int* in_sizes, int n_in,
                              void* d_out, int out_size, void* d_ws, size_t ws_size,
                              hipStream_t stream) {
  (void)in_sizes; (void)n_in; (void)out_size; (void)ws_size;
  const float* img   = (const float*)d_in[0];
  const float* p0    = (const float*)d_in[1];
  const float* p1    = (const float*)d_in[2];
  const float* p2    = (const float*)d_in[3];
  const float* rp    = (const float*)d_in[4];
  const float* boxes = (const float*)d_in[5];
  float* out = (float*)d_out;

  char* ws = (char*)d_ws;
  unsigned*           mask   = (unsigned*)(ws + WS_MASK);
  unsigned*           keep   = (unsigned*)(ws + WS_KEEP);
  unsigned long long* keys   = (unsigned long long*)(ws + WS_KEYS);
  unsigned*           order  = (unsigned*)(ws + WS_ORDER);
  float* x1s   = (float*)(ws + WS_X1S);
  float* y1s   = (float*)(ws + WS_Y1S);
  float* x2s   = (float*)(ws + WS_X2S);
  float* y2s   = (float*)(ws + WS_Y2S);
  float* area  = (float*)(ws + WS_AREA);
  float* confs = (float*)(ws + WS_CONFS);
  float* c5s   = (float*)(ws + WS_C5S);
  float* ppart  = (float*)(ws + WS_PPART);
  float* rpart  = (float*)(ws + WS_RPART);
  float* blpart = (float*)(ws + WS_BLPART);
  float* bcpart = (float*)(ws + WS_BCPART);

  // mask + keep are contiguous: clear both in one pass
  int clearN = M_BOXES * MASK_PITCH + MASK_PITCH;
  clear_k<<<(clearN + 255) / 256, 256, 0, stream>>>(mask, clearN);

  patch_loss_k<<<P_BLOCKS, 256, 0, stream>>>(img, p0, p1, p2, ppart);
  rcnn_k<<<R_BLOCKS, 256, 0, stream>>>(rp, rpart);
  yolo_prep_k<<<YP_BLOCKS, 256, 0, stream>>>(boxes, keys, blpart, bcpart);
  bitonic_k<<<1, 1024, SORT_N * sizeof(unsigned long long), stream>>>(keys, order);
  gather_k<<<R_BLOCKS, 256, 0, stream>>>(boxes, order, x1s, y1s, x2s, y2s, area, confs, c5s);

  dim3 g(M_BOXES / 16, M_BOXES / 16);   // 384 x 384 tiles
  nms_mask_k<<<g, 32, 0, stream>>>(x1s, y1s, x2s, y2s, area, mask);
  nms_scan_k<<<1, 32, 0, stream>>>(mask, confs, keep);
  final_k<<<1, 256, 0, stream>>>(ppart, rpart, blpart, bcpart, keep, confs, c5s, out);
}